// CostVolumeLayer_87995289960694
// MI455X (gfx1250) — compile-verified
//
#include <hip/hip_runtime.h>
#include <hip/hip_bf16.h>

// Cost volume via banded-Gram FP32 WMMA (V_WMMA_F32_16X16X4_F32).
// out[b,s,h,w] = sum_c src[b,c,h,w] * tgt_padded[b,c,h+dh_s,w+dw_s]
// s order: [ (0,0), then for i=1..4: (-i,0),(i,0),(0,-i),(0,i) ]

typedef float v2f __attribute__((ext_vector_type(2)));
typedef float v8f __attribute__((ext_vector_type(8)));

constexpr int BB = 4;
constexpr int C  = 128;
constexpr int H  = 96;
constexpr int W  = 160;
constexpr int R  = 4;
constexpr int S  = 1 + 4 * R;   // 17
constexpr int HW = H * W;

static_assert(W % 16 == 0, "W tileable");
static_assert(H % 16 == 0, "H tileable");
static_assert(C % 4 == 0, "C divisible by WMMA K");

// ---------------------------------------------------------------------------
// W-pass: 16-pixel tiles along W. Gram of src tile (16xC) with tgt window
// (cols w0-8 .. w0+23) -> diagonals d in [-4,4] give identity + horizontal
// shifts (s = 0, and 4(i-1)+3 / 4(i-1)+4).
// One wave (32 threads) per tile.
// ---------------------------------------------------------------------------
__global__ __launch_bounds__(32)
void cost_volume_wpass(const float* __restrict__ src,
                       const float* __restrict__ tgt,
                       float* __restrict__ out) {
    const int WT  = W / 16;
    int id  = blockIdx.x;                 // b * (H*WT) + h * WT + tw
    int b   = id / (H * WT);
    int rem = id % (H * WT);
    int h   = rem / WT;
    int w0  = (rem % WT) * 16;

    int lane  = threadIdx.x;              // 0..31, wave32
    int n     = lane & 15;                // M for A, N for B, N for D
    int khalf = lane >> 4;                // K-half selector (K = 2*khalf + reg)

    const float* srcb = src + (size_t)b * C * HW;
    const float* tgtb = tgt + (size_t)b * C * HW;

    // A: src[c, h, w0+n]   (row-major pixels along W -> coalesced)
    const int aoff = h * W + w0 + n;

    // B chains: window columns [w0-8, w0+7] and [w0+8, w0+23], zero-pad OOB.
    int col0 = w0 - 8 + n;
    int col1 = w0 + 8 + n;
    bool ok0 = (col0 >= 0) && (col0 < W);
    bool ok1 = (col1 >= 0) && (col1 < W);
    int b0off = h * W + min(max(col0, 0), W - 1);
    int b1off = h * W + min(max(col1, 0), W - 1);

    v8f acc0 = {};
    v8f acc1 = {};

    for (int c = 0; c < C; c += 4) {
        int cA = c + 2 * khalf;
        size_t o0 = (size_t)cA * HW;
        size_t o1 = (size_t)(cA + 1) * HW;

        v2f a, b0, b1;
        a.x = srcb[o0 + aoff];
        a.y = srcb[o1 + aoff];

        float t00 = tgtb[o0 + b0off];
        float t01 = tgtb[o1 + b0off];
        float t10 = tgtb[o0 + b1off];
        float t11 = tgtb[o1 + b1off];
        b0.x = ok0 ? t00 : 0.0f;  b0.y = ok0 ? t01 : 0.0f;
        b1.x = ok1 ? t10 : 0.0f;  b1.y = ok1 ? t11 : 0.0f;

        if (c + 4 < C)  // speculative prefetch of next K-step's src line
            __builtin_prefetch(&srcb[(size_t)(cA + 4) * HW + aoff], 0, 1);

        acc0 = __builtin_amdgcn_wmma_f32_16x16x4_f32(
            false, a, false, b0, (short)0, acc0, false, false);
        acc1 = __builtin_amdgcn_wmma_f32_16x16x4_f32(
            false, a, false, b1, (short)0, acc1, false, false);
    }

    // De-swizzle D through LDS: D reg r, lane l -> (M = r + 8*(l>>4), N = l&15).
    __shared__ float smem[16][33];
    #pragma unroll
    for (int r = 0; r < 8; ++r) {
        smem[r + 8 * khalf][n]      = acc0[r];
        smem[r + 8 * khalf][n + 16] = acc1[r];
    }
    __syncthreads();

    if (lane < 16) {
        int p = lane;  // pixel within tile
        #pragma unroll
        for (int d = -R; d <= R; ++d) {
            int s = (d == 0) ? 0 : (d < 0 ? 4 * (-d - 1) + 3 : 4 * (d - 1) + 4);
            out[(((size_t)b * S + s) * H + h) * W + w0 + p] = smem[p][p + d + 8];
        }
    }
}

// ---------------------------------------------------------------------------
// H-pass: 16-pixel tiles along H (fixed w). Gram of src tile with tgt rows
// h0-8 .. h0+23 -> diagonals d in [-4,4]\{0} give vertical shifts
// (s = 4(i-1)+1 / 4(i-1)+2). Identity already written by the W-pass.
// ---------------------------------------------------------------------------
__global__ __launch_bounds__(32)
void cost_volume_hpass(const float* __restrict__ src,
                       const float* __restrict__ tgt,
                       float* __restrict__ out) {
    const int HT  = H / 16;
    int id  = blockIdx.x;                 // b * (W*HT) + w * HT + th
    int b   = id / (W * HT);
    int rem = id % (W * HT);
    int w   = rem / HT;
    int h0  = (rem % HT) * 16;

    int lane  = threadIdx.x;
    int n     = lane & 15;
    int khalf = lane >> 4;

    const float* srcb = src + (size_t)b * C * HW;
    const float* tgtb = tgt + (size_t)b * C * HW;

    // A: src[c, h0+n, w]
    const int aoff = (h0 + n) * W + w;

    // B chains: window rows [h0-8, h0+7] and [h0+8, h0+23], zero-pad OOB.
    int row0 = h0 - 8 + n;
    int row1 = h0 + 8 + n;
    bool ok0 = (row0 >= 0) && (row0 < H);
    bool ok1 = (row1 >= 0) && (row1 < H);
    int b0off = min(max(row0, 0), H - 1) * W + w;
    int b1off = min(max(row1, 0), H - 1) * W + w;

    v8f acc0 = {};
    v8f acc1 = {};

    for (int c = 0; c < C; c += 4) {
        int cA = c + 2 * khalf;
        size_t o0 = (size_t)cA * HW;
        size_t o1 = (size_t)(cA + 1) * HW;

        v2f a, b0, b1;
        a.x = srcb[o0 + aoff];
        a.y = srcb[o1 + aoff];

        float t00 = tgtb[o0 + b0off];
        float t01 = tgtb[o1 + b0off];
        float t10 = tgtb[o0 + b1off];
        float t11 = tgtb[o1 + b1off];
        b0.x = ok0 ? t00 : 0.0f;  b0.y = ok0 ? t01 : 0.0f;
        b1.x = ok1 ? t10 : 0.0f;  b1.y = ok1 ? t11 : 0.0f;

        if (c + 4 < C)
            __builtin_prefetch(&srcb[(size_t)(cA + 4) * HW + aoff], 0, 1);

        acc0 = __builtin_amdgcn_wmma_f32_16x16x4_f32(
            false, a, false, b0, (short)0, acc0, false, false);
        acc1 = __builtin_amdgcn_wmma_f32_16x16x4_f32(
            false, a, false, b1, (short)0, acc1, false, false);
    }

    __shared__ float smem[16][33];
    #pragma unroll
    for (int r = 0; r < 8; ++r) {
        smem[r + 8 * khalf][n]      = acc0[r];
        smem[r + 8 * khalf][n + 16] = acc1[r];
    }
    __syncthreads();

    if (lane < 16) {
        int p = lane;  // row within tile
        #pragma unroll
        for (int d = -R; d <= R; ++d) {
            if (d == 0) continue;  // identity handled by W-pass
            int s = (d < 0) ? 4 * (-d - 1) + 1 : 4 * (d - 1) + 2;
            out[(((size_t)b * S + s) * H + (h0 + p)) * W + w] = smem[p][p + d + 8];
        }
    }
}

extern "C" void kernel_launch(void* const* d_in, const int* in_sizes, int n_in,
                              void* d_out, int out_size, void* d_ws, size_t ws_size,
                              hipStream_t stream) {
    const float* src = (const float*)d_in[0];
    const float* tgt = (const float*)d_in[1];
    float* out = (float*)d_out;

    const int grid_w = BB * H * (W / 16);   // 3840 waves
    const int grid_h = BB * W * (H / 16);   // 3840 waves

    cost_volume_wpass<<<grid_w, 32, 0, stream>>>(src, tgt, out);
    cost_volume_hpass<<<grid_h, 32, 0, stream>>>(src, tgt, out);
}